// SimpleGNN_1434519077392
// MI455X (gfx1250) — compile-verified
//
#include <hip/hip_runtime.h>
#include <hip/hip_bf16.h>

typedef float v2f __attribute__((ext_vector_type(2)));
typedef float v8f __attribute__((ext_vector_type(8)));

#define F_IN   128
#define HID    128
#define OUTF   64
#define WAVES_PER_BLOCK 4
#define T_STRIDE 132   // padded row stride (floats) for LDS tile: avoids bank conflicts

// Force the native global_atomic_add_f32 (no CAS loop): destination is plain
// hipMalloc'd workspace, so the "unsafe" (no fine-grained memory) assumption holds.
__device__ __forceinline__ void atomic_add_f32(float* p, float v) {
#if defined(__HIP_DEVICE_COMPILE__)
    unsafeAtomicAdd(p, v);
#else
    __hip_atomic_fetch_add(p, v, __ATOMIC_RELAXED, __HIP_MEMORY_SCOPE_AGENT);
#endif
}

// ---------------------------------------------------------------------------
// Encoder: h = relu(X @ W1 + b1) @ W2 + b2 ; also hfinal = h (scatter seed)
// One wave32 per 16-row block. fp32 WMMA 16x16x4.
// ---------------------------------------------------------------------------
__global__ void __launch_bounds__(32 * WAVES_PER_BLOCK)
gnn_encoder(const float* __restrict__ X,
            const float* __restrict__ W1, const float* __restrict__ b1,
            const float* __restrict__ W2, const float* __restrict__ b2,
            float* __restrict__ h, float* __restrict__ hfinal, int nNodes)
{
    __shared__ float tbuf[WAVES_PER_BLOCK][16 * T_STRIDE];

    const int lane  = threadIdx.x & 31;
    const int wave  = threadIdx.x >> 5;
    const int lhalf = lane >> 4;   // 0: K+0/K+1 rows 0..15 ; 1: K+2/K+3 rows 0..15
    const int lmod  = lane & 15;
    float* t = tbuf[wave];

    const int nrb = nNodes >> 4;   // 16-row blocks
    for (int rb = blockIdx.x * WAVES_PER_BLOCK + wave; rb < nrb;
         rb += gridDim.x * WAVES_PER_BLOCK) {
        const int row0 = rb << 4;

        // ---------------- GEMM1: acc = X[row0:row0+16, :] @ W1 ----------------
        v8f acc[8] = {};
        {
            // A-frag: lane l<16 -> row row0+l, cols {k,k+1}; l>=16 -> row row0+l-16, cols {k+2,k+3}
            const float* xrow = X + (size_t)(row0 + lmod) * F_IN + lhalf * 2;
            for (int ks = 0; ks < F_IN / 4; ++ks) {
                v2f a = *(const v2f*)(xrow + ks * 4);
                const int K0 = ks * 4 + lhalf * 2;
#pragma unroll
                for (int nt = 0; nt < 8; ++nt) {
                    const int N = nt * 16 + lmod;
                    v2f bf;
                    bf.x = W1[(size_t)K0 * HID + N];
                    bf.y = W1[(size_t)(K0 + 1) * HID + N];
                    acc[nt] = __builtin_amdgcn_wmma_f32_16x16x4_f32(
                        false, a, false, bf, (short)0, acc[nt], false, false);
                }
            }
        }

        // bias + ReLU, transpose C-layout -> row-major LDS tile (A source for GEMM2)
#pragma unroll
        for (int nt = 0; nt < 8; ++nt) {
            const int N = nt * 16 + lmod;
            const float bias = b1[N];
#pragma unroll
            for (int r = 0; r < 8; ++r) {
                float v = acc[nt][r] + bias;
                v = v > 0.f ? v : 0.f;
                t[(r + lhalf * 8) * T_STRIDE + N] = v;
            }
        }

        // ---------------- GEMM2: acc2 = t @ W2 ----------------
        v8f acc2[8] = {};
        {
            const float* trow = t + lmod * T_STRIDE + lhalf * 2;
            for (int ks = 0; ks < HID / 4; ++ks) {
                v2f a = *(const v2f*)(trow + ks * 4);   // ds_load_b64, same-wave ordering
                const int K0 = ks * 4 + lhalf * 2;
#pragma unroll
                for (int nt = 0; nt < 8; ++nt) {
                    const int N = nt * 16 + lmod;
                    v2f bf;
                    bf.x = W2[(size_t)K0 * HID + N];
                    bf.y = W2[(size_t)(K0 + 1) * HID + N];
                    acc2[nt] = __builtin_amdgcn_wmma_f32_16x16x4_f32(
                        false, a, false, bf, (short)0, acc2[nt], false, false);
                }
            }
        }

        // bias + store to h and hfinal (lanes 0..15 of a tile cover 64 contiguous bytes)
#pragma unroll
        for (int nt = 0; nt < 8; ++nt) {
            const int N = nt * 16 + lmod;
            const float bias = b2[N];
#pragma unroll
            for (int r = 0; r < 8; ++r) {
                const float v = acc2[nt][r] + bias;
                const size_t idx = (size_t)(row0 + r + lhalf * 8) * HID + N;
                h[idx]      = v;
                hfinal[idx] = v;
            }
        }
    }
}

// ---------------------------------------------------------------------------
// Scatter: hfinal[dst] += h[src], one wave per edge, float4 per lane
// ---------------------------------------------------------------------------
__global__ void __launch_bounds__(256)
gnn_scatter(const float* __restrict__ h, const int* __restrict__ ei,
            float* __restrict__ hfinal, int nEdges)
{
    const long long gid = (long long)blockIdx.x * blockDim.x + threadIdx.x;
    const long long e   = gid >> 5;
    if (e >= nEdges) return;
    const int lane = (int)(gid & 31);

    const int src = ei[e];
    const int dst = ei[(size_t)nEdges + e];

    const float4 v = ((const float4*)h)[(size_t)src * 32 + lane];
    float* out = hfinal + (size_t)dst * HID + lane * 4;
    atomic_add_f32(out + 0, v.x);
    atomic_add_f32(out + 1, v.y);
    atomic_add_f32(out + 2, v.z);
    atomic_add_f32(out + 3, v.w);
}

// ---------------------------------------------------------------------------
__global__ void gnn_zero(float* __restrict__ pooled)
{
    pooled[threadIdx.x] = 0.f;
}

// Column sums of hfinal into pooled[128]
__global__ void __launch_bounds__(256)
gnn_reduce(const float* __restrict__ hfinal, float* __restrict__ pooled, int nNodes)
{
    const int col  = threadIdx.x & 127;
    const int half = threadIdx.x >> 7;
    float s = 0.f;
    for (int row = blockIdx.x * 2 + half; row < nNodes; row += gridDim.x * 2)
        s += hfinal[(size_t)row * HID + col];
    atomic_add_f32(&pooled[col], s);
}

// out = (pooled / N) @ Wp + bp
__global__ void __launch_bounds__(64)
gnn_head(const float* __restrict__ pooled, const float* __restrict__ Wp,
         const float* __restrict__ bp, float* __restrict__ out, int nNodes)
{
    const int o = threadIdx.x;
    const float invN = 1.0f / (float)nNodes;
    float acc = 0.f;
#pragma unroll 4
    for (int j = 0; j < HID; ++j)
        acc += pooled[j] * Wp[(size_t)j * OUTF + o];
    out[o] = acc * invN + bp[o];
}

// ---------------------------------------------------------------------------
extern "C" void kernel_launch(void* const* d_in, const int* in_sizes, int n_in,
                              void* d_out, int out_size, void* d_ws, size_t ws_size,
                              hipStream_t stream)
{
    const float* X  = (const float*)d_in[0];
    const int*   ei = (const int*)  d_in[1];   // jax default x64-off => int32
    const float* W1 = (const float*)d_in[2];
    const float* b1 = (const float*)d_in[3];
    const float* W2 = (const float*)d_in[4];
    const float* b2 = (const float*)d_in[5];
    const float* Wp = (const float*)d_in[6];
    const float* bp = (const float*)d_in[7];
    float* out = (float*)d_out;

    const int nNodes = in_sizes[0] / F_IN;   // 100000
    const int nEdges = in_sizes[1] / 2;      // 1000000

    float* h      = (float*)d_ws;
    float* hfinal = h + (size_t)nNodes * HID;
    float* pooled = hfinal + (size_t)nNodes * HID;

    // 1) encoder
    const int nrb = nNodes / 16;
    const int encBlocks = (nrb + WAVES_PER_BLOCK - 1) / WAVES_PER_BLOCK;
    gnn_encoder<<<encBlocks, 32 * WAVES_PER_BLOCK, 0, stream>>>(
        X, W1, b1, W2, b2, h, hfinal, nNodes);

    // 2) zero pooled accumulator (must happen before reduce; stream-ordered)
    gnn_zero<<<1, HID, 0, stream>>>(pooled);

    // 3) edge scatter-add
    const long long totThreads = (long long)nEdges * 32;
    const int sBlocks = (int)((totThreads + 255) / 256);
    gnn_scatter<<<sBlocks, 256, 0, stream>>>(h, ei, hfinal, nEdges);

    // 4) column mean numerator
    gnn_reduce<<<1024, 256, 0, stream>>>(hfinal, pooled, nNodes);

    // 5) head
    gnn_head<<<1, OUTF, 0, stream>>>(pooled, Wp, bp, out, nNodes);
}